// SuperPixelSamplingNet_59742995088000
// MI455X (gfx1250) — compile-verified
//
#include <hip/hip_runtime.h>

// Problem constants (match reference)
#define Hn   1024
#define Wn   1024
#define HWn  (1024*1024)
#define Bn   2
#define Cn   5
#define KHn  32
#define KWn  32
#define Kn   (KHn*KWn)
#define NITER 10
#define EPSf  1e-10f
#define NEGBIG 1e16f

typedef float v2f __attribute__((ext_vector_type(2)));
typedef float v8f __attribute__((ext_vector_type(8)));
typedef unsigned int v4u __attribute__((ext_vector_type(4)));
typedef int v4i __attribute__((ext_vector_type(4)));
typedef int v8i __attribute__((ext_vector_type(8)));

// ---------------------------------------------------------------------------
// Init: sp[b,k,c] = mean of x over the 32x32 tile of grid cell k.
// ---------------------------------------------------------------------------
__global__ __launch_bounds__(256) void init_kernel(const float* __restrict__ x,
                                                   float* __restrict__ sp) {
  __shared__ float s[Cn];
  int t = threadIdx.x;
  if (t < Cn) s[t] = 0.f;
  __syncthreads();
  int blk  = blockIdx.x;
  int b    = blk >> 10;
  int cell = blk & 1023;
  int y0 = (cell >> 5) * 32, x0 = (cell & 31) * 32;
  float part[Cn];
#pragma unroll
  for (int c = 0; c < Cn; ++c) part[c] = 0.f;
  for (int p = t; p < 1024; p += 256) {
    int py = p >> 5, px = p & 31;
    size_t base = (size_t)(y0 + py) * Wn + (x0 + px);
#pragma unroll
    for (int c = 0; c < Cn; ++c)
      part[c] += x[((size_t)(b * Cn + c)) * HWn + base];
  }
#pragma unroll
  for (int c = 0; c < Cn; ++c) atomicAdd(&s[c], part[c]);
  __syncthreads();
  if (t < Cn)
    sp[((size_t)(b * Kn + cell)) * Cn + t] = s[t] * (1.0f / 1024.0f);
}

// ---------------------------------------------------------------------------
// Zero scratch accumulators
// ---------------------------------------------------------------------------
__global__ void zero_kernel(float* __restrict__ p, int n) {
  int i = blockIdx.x * 256 + threadIdx.x;
  if (i < n) p[i] = 0.f;
}

// ---------------------------------------------------------------------------
// Assign + accumulate. One block per (b, cell). 256 threads = 8 waves.
// Tile staged to LDS by the Tensor Data Mover (one 32x32 f32 tile per
// channel; 4KB row stride in global, contiguous in LDS), overlapped with
// scalar staging of the 9 neighbor centers, completed by s_wait_tensorcnt.
// Dot products pixel·center via V_WMMA_F32_16X16X4_F32 (K=5 padded to 8,
// two chained WMMAs). softmax(2*dot - |g|^2 - invis*1e16) over 9 candidates.
// Weighted sums privatized into 54 LDS accumulators, flushed with atomics.
// final_pass: also writes hard-argmax superpixel index (int ws + float out).
// ---------------------------------------------------------------------------
__global__ __launch_bounds__(256) void assign_kernel(
    const float* __restrict__ x, const float* __restrict__ sp,
    float* __restrict__ num, float* __restrict__ den,
    int* __restrict__ idx_out, float* __restrict__ idxf_out, int final_pass) {
  __shared__ __align__(16) float ldsF[Cn][1024]; // tile features, channel-major
  __shared__ float ldsB[8][16];       // B matrix: [k][n] = center n, channel k
  __shared__ float ldsAdj[16];        // -|g_n|^2 - invis_n*NEG_BIG
  __shared__ int   ldsSp[16];         // superpixel id of candidate n
  __shared__ float ldsDot[8][16][17]; // per-wave D tile (padded vs bank conflicts)
  __shared__ float ldsAcc[9][6];      // per-block partial sums (5 num + 1 den)

  int t    = threadIdx.x;
  int blk  = blockIdx.x;
  int b    = blk >> 10;
  int cell = blk & 1023;
  int cy = cell >> 5, cx = cell & 31;
  int y0 = cy * 32, x0 = cx * 32;

  // --- TDM: DMA the 32x32 tile of each channel into LDS (wave 0 issues). ---
  if (t == 0) {
#pragma unroll
    for (int c = 0; c < Cn; ++c) {
      unsigned long long ga = (unsigned long long)(uintptr_t)(
          x + ((size_t)(b * Cn + c)) * HWn + (size_t)y0 * Wn + x0);
      unsigned int lds_off = (unsigned int)(uintptr_t)&ldsF[c][0];  // LDS offset
      v4u g0;
      g0.x = 1u;                                        // count=1 (valid D#)
      g0.y = lds_off;                                   // lds_addr[31:0]
      g0.z = (unsigned int)(ga & 0xffffffffu);          // global_addr[31:0]
      g0.w = (unsigned int)((ga >> 32) & 0x01ffffffu)   // global_addr[56:32]
             | (2u << 30);                              // type=2 ("image")
      v8i g1;
      g1[0] = 0x20000;                 // data_size=2 (4B); mask/flags = 0
      g1[1] = (int)(1024u << 16);      // tensor_dim0 = 1024 (low 16 in [31:16])
      g1[2] = (int)(1024u << 16);      // dim0 hi16=0 | tensor_dim1 = 1024
      g1[3] = (int)(32u << 16);        // dim1 hi16=0 | tile_dim0 = 32
      g1[4] = 32;                      // tile_dim1 = 32, tile_dim2 = 0
      g1[5] = 1024;                    // tensor_dim0_stride = 1024 elements
      g1[6] = 0;                       // stride hi | tensor_dim1_stride lo = 0
      g1[7] = 0;
      v4i z4 = {0, 0, 0, 0};           // groups 2/3 unused for 2D tensors
      v8i z8 = {0, 0, 0, 0, 0, 0, 0, 0};
      __builtin_amdgcn_tensor_load_to_lds(g0, g1, z4, z4, z8, 0);
    }
  }

  // --- Concurrently stage the 9 neighbor centers as zero-padded 8x16 B. ---
  if (t < 128) {
    int k = t >> 4, n = t & 15;
    float v = 0.f;
    if (n < 9) {
      int dy = n / 3 - 1, dx = n % 3 - 1;
      int ny = cy + dy, nx = cx + dx;
      int cny = min(max(ny, 0), KHn - 1);
      int cnx = min(max(nx, 0), KWn - 1);
      int s = cny * KWn + cnx;
      if (k == 0) ldsSp[n] = s;
      if (k < Cn) v = sp[((size_t)(b * Kn + s)) * Cn + k];
    } else if (k == 0) {
      ldsSp[n] = 0;
    }
    ldsB[k][n] = v;
  }
  if (t < 54) ldsAcc[t / 6][t % 6] = 0.f;
  if (t == 0) __builtin_amdgcn_s_wait_tensorcnt(0);  // tile DMA complete
  __syncthreads();
  if (t < 16) {
    float adj = -3.0e38f;
    if (t < 9) {
      float g2 = 0.f;
#pragma unroll
      for (int k = 0; k < Cn; ++k) { float g = ldsB[k][t]; g2 += g * g; }
      int dy = t / 3 - 1, dx = t % 3 - 1;
      int ny = cy + dy, nx = cx + dx;
      bool bad = (ny < 0) | (ny >= KHn) | (nx < 0) | (nx >= KWn);
      adj = -g2 - (bad ? NEGBIG : 0.f);
    }
    ldsAdj[t] = adj;
  }
  __syncthreads();

  int w = t >> 5, lane = t & 31;
  int half = lane >> 4, mn = lane & 15;

  float acc[9][6];
#pragma unroll
  for (int n = 0; n < 9; ++n)
#pragma unroll
    for (int c = 0; c < 6; ++c) acc[n][c] = 0.f;

  // B fragments (ISA layout: V0 holds K=2*half, V1 holds K=2*half+1; col = lane&15)
  v2f bf0, bf1;
  bf0.x = ldsB[half * 2 + 0][mn];
  bf0.y = ldsB[half * 2 + 1][mn];
  bf1.x = ldsB[4 + half * 2 + 0][mn];
  bf1.y = ldsB[4 + half * 2 + 1][mn];

  for (int g = 0; g < 8; ++g) {        // each wave: 8 groups of 16 pixels
    int group = w * 8 + g;
    int pixA = group * 16 + mn;
    // A fragments: lanes 0-15 carry channels (0,1), lanes 16-31 carry (2,3);
    // second WMMA carries (4,pad) / (pad,pad).
    v2f af0, af1;
    af0.x = ldsF[half * 2 + 0][pixA];
    af0.y = ldsF[half * 2 + 1][pixA];
    af1.x = half ? 0.f : ldsF[4][pixA];
    af1.y = 0.f;

    v8f d = {0.f, 0.f, 0.f, 0.f, 0.f, 0.f, 0.f, 0.f};
    d = __builtin_amdgcn_wmma_f32_16x16x4_f32(false, af0, false, bf0,
                                              (short)0, d, false, false);
    d = __builtin_amdgcn_wmma_f32_16x16x4_f32(false, af1, false, bf1,
                                              (short)0, d, false, false);
    // D layout: VGPR j -> row j (lanes 0-15) / row j+8 (lanes 16-31), col = lane&15
#pragma unroll
    for (int j = 0; j < 8; ++j) ldsDot[w][j + half * 8][mn] = d[j];
    __syncthreads();

    if (half == 0) {  // lanes 0-15: one pixel each, full softmax over 9
      int pix = group * 16 + mn;
      float lg[9], mx = -3.0e38f;
#pragma unroll
      for (int n = 0; n < 9; ++n) {
        lg[n] = 2.0f * ldsDot[w][mn][n] + ldsAdj[n];
        mx = fmaxf(mx, lg[n]);
      }
      float e[9], ssum = 0.f;
#pragma unroll
      for (int n = 0; n < 9; ++n) { e[n] = __expf(lg[n] - mx); ssum += e[n]; }
      float inv = 1.0f / ssum;
      float f0 = ldsF[0][pix], f1 = ldsF[1][pix], f2 = ldsF[2][pix];
      float f3 = ldsF[3][pix], f4 = ldsF[4][pix];
#pragma unroll
      for (int n = 0; n < 9; ++n) {
        float wn = e[n] * inv;
        acc[n][0] += wn * f0; acc[n][1] += wn * f1; acc[n][2] += wn * f2;
        acc[n][3] += wn * f3; acc[n][4] += wn * f4; acc[n][5] += wn;
      }
      if (final_pass) {
        int best = 0; float bv = lg[0];
#pragma unroll
        for (int n = 1; n < 9; ++n)
          if (lg[n] > bv) { bv = lg[n]; best = n; }
        int py = pix >> 5, px = pix & 31;
        size_t gi = (size_t)b * HWn + (size_t)(y0 + py) * Wn + (x0 + px);
        int sid = ldsSp[best];
        idx_out[gi]  = sid;
        idxf_out[gi] = (float)sid;
      }
    }
    __syncthreads();
  }

  // Merge register partials into LDS, then 54 global atomics per block.
  if (half == 0) {
#pragma unroll
    for (int n = 0; n < 9; ++n)
#pragma unroll
      for (int c = 0; c < 6; ++c) atomicAdd(&ldsAcc[n][c], acc[n][c]);
  }
  __syncthreads();
  if (t < 54) {
    int n = t / 6, c = t % 6;
    int s = ldsSp[n];
    float v = ldsAcc[n][c];
    if (c < Cn) atomicAdd(&num[((size_t)(b * Kn + s)) * Cn + c], v);
    else        atomicAdd(&den[b * Kn + s], v);
  }
}

// ---------------------------------------------------------------------------
// sp = num / (den + eps)
// ---------------------------------------------------------------------------
__global__ void normalize_kernel(const float* __restrict__ num,
                                 const float* __restrict__ den,
                                 float* __restrict__ sp) {
  int i = blockIdx.x * 256 + threadIdx.x;
  if (i < Bn * Kn * Cn) sp[i] = num[i] / (den[i / Cn] + EPSf);
}

// ---------------------------------------------------------------------------
// recon[b,c,h,w] = new_sp[b, idx[b,h,w], c]
// ---------------------------------------------------------------------------
__global__ void gather_kernel(const float* __restrict__ sp,
                              const int* __restrict__ idx,
                              float* __restrict__ out) {
  int i = blockIdx.x * 256 + threadIdx.x;
  if (i >= Bn * HWn) return;
  int b = i / HWn;
  int p = i - b * HWn;
  int s = idx[i];
  const float* spb = sp + ((size_t)(b * Kn + s)) * Cn;
#pragma unroll
  for (int c = 0; c < Cn; ++c)
    out[((size_t)(b * Cn + c)) * HWn + p] = spb[c];
}

// ---------------------------------------------------------------------------
// Host launch
// ---------------------------------------------------------------------------
extern "C" void kernel_launch(void* const* d_in, const int* in_sizes, int n_in,
                              void* d_out, int out_size, void* d_ws, size_t ws_size,
                              hipStream_t stream) {
  (void)in_sizes; (void)n_in; (void)out_size; (void)ws_size;
  const float* x = (const float*)d_in[0];
  float* out = (float*)d_out;

  // Workspace layout (floats): sp[10240] | num[10240] | den[2048] | idx[B*HW as int]
  float* wsf = (float*)d_ws;
  float* sp  = wsf;
  float* num = wsf + Bn * Kn * Cn;           // 10240
  float* den = num + Bn * Kn * Cn;           // +10240
  int*   idx = (int*)(den + Bn * Kn);        // +2048
  float* idxf_out = out + (size_t)Bn * Cn * HWn;

  const int nBlocks = Bn * Kn;               // 2048: one per (b, grid cell)
  const int nAccum  = Bn * Kn * Cn + Bn * Kn; // num+den contiguous: 12288

  init_kernel<<<nBlocks, 256, 0, stream>>>(x, sp);

  for (int it = 0; it < NITER; ++it) {
    zero_kernel<<<(nAccum + 255) / 256, 256, 0, stream>>>(num, nAccum);
    assign_kernel<<<nBlocks, 256, 0, stream>>>(x, sp, num, den, idx, idxf_out, 0);
    normalize_kernel<<<(Bn * Kn * Cn + 255) / 256, 256, 0, stream>>>(num, den, sp);
  }

  // Final association: accumulate with raw x weights + hard argmax indices.
  zero_kernel<<<(nAccum + 255) / 256, 256, 0, stream>>>(num, nAccum);
  assign_kernel<<<nBlocks, 256, 0, stream>>>(x, sp, num, den, idx, idxf_out, 1);
  normalize_kernel<<<(Bn * Kn * Cn + 255) / 256, 256, 0, stream>>>(num, den, sp);
  gather_kernel<<<(Bn * HWn + 255) / 256, 256, 0, stream>>>(sp, idx, out);
}